// RPN_20658792693916
// MI455X (gfx1250) — compile-verified
//
#include <hip/hip_runtime.h>
#include <hip/hip_bf16.h>
#include <math.h>

typedef __attribute__((ext_vector_type(2))) float v2f;
typedef __attribute__((ext_vector_type(8))) float v8f;
typedef __attribute__((ext_vector_type(4))) unsigned int v4u;
typedef __attribute__((ext_vector_type(8))) int v8i;
typedef __attribute__((ext_vector_type(4))) int v4i;

#define BSZ   16
#define CIN   512
#define FC    256
#define HH    32
#define WW    32
#define NPIX  1024          // 32*32
#define ACNT  9
#define NANCH 9216          // 1024*9
#define NPAD  16384
#define MAXOUT 256
#define NMSTHR 0.7f

// ---------------- K0a: NCHW -> NHWC feature transpose ----------------
__global__ void k_transpose_feat(const float* __restrict__ feat, float* __restrict__ featT) {
    int idx = blockIdx.x * blockDim.x + threadIdx.x;   // b*512*1024 + c*1024 + p
    if (idx >= BSZ * CIN * NPIX) return;
    int p = idx & (NPIX - 1);
    int c = (idx >> 10) & (CIN - 1);
    int b = idx >> 19;
    featT[((size_t)(b * NPIX + p)) * CIN + c] = feat[idx];
}

// ---------------- K0b: zero pad buffer ----------------
__global__ void k_zero(float* __restrict__ z) {
    z[threadIdx.x] = 0.0f;
}

// ---------------- K1: weight repack OIHW -> [r][c/4][n][c%4] ----------------
__global__ void k_repack_w(const float* __restrict__ w1, float* __restrict__ wT2) {
    int idx = blockIdx.x * blockDim.x + threadIdx.x;
    if (idx >= 9 * CIN * FC) return;
    int j  = idx & 3;
    int n  = (idx >> 2) & 255;
    int cb = (idx >> 10) & 127;
    int r  = idx >> 17;
    int c  = cb * 4 + j;
    wT2[idx] = w1[(size_t)n * (CIN * 9) + c * 9 + r];
}

// ---------------- K2: conv3x3 + bias + ReLU, f32 WMMA + TDM-staged weights ----------------
// 256-thread workgroup = 8 waves (4 Mtiles x 2 Ntiles) -> 64x32 output tile.
// Weights staged to LDS by the Tensor Data Mover, double-buffered over 36 K-chunks.
__global__ void __launch_bounds__(256)
k_conv_wmma(const float* __restrict__ featT,
            const float* __restrict__ zbuf,
            const float* __restrict__ wT2,
            const float* __restrict__ b1,
            float* __restrict__ Lws) {
    __shared__ float Bs[2][32 * 128];     // 2 x 16KB chunks: [cb_local][nn*4+j]

    int bid = blockIdx.x;                 // 16 batches * 16 mgroups * 8 ngroups
    int b  = bid >> 7;
    int mg = (bid >> 3) & 15;
    int ng = bid & 7;

    int tid  = threadIdx.x;
    int wv   = tid >> 5;
    int lane = tid & 31;
    int lm   = lane & 15;
    int kb   = (lane >> 4) * 2;           // K sub-offset {0,2}
    int mtl  = wv & 3;                    // wave's M tile within group
    int ntl  = wv >> 2;                   // wave's N tile within group (0..1)

    int m0 = mg * 64 + mtl * 16;
    int m  = m0 + lm;
    int py = m >> 5, px = m & 31;
    int n0 = ng * 32 + ntl * 16;
    int nn = ntl * 16 + lm;               // column within 32-wide LDS slab

    const float* fT = featT + (size_t)b * NPIX * CIN;
    v8f acc = {0.f, 0.f, 0.f, 0.f, 0.f, 0.f, 0.f, 0.f};

    // --- TDM issue helper: load one 32(cb) x 128(floats) weight chunk to LDS ---
    auto issue_tdm = [&](int c, int buf) {
        int r  = c >> 2;
        int ch = c & 3;
        const float* gp = wT2 + (size_t)r * (128 * 1024) + (size_t)ch * (32 * 1024) + ng * 128;
        unsigned long long ga = (unsigned long long)(uintptr_t)gp;
        unsigned ldsa = (unsigned)(uintptr_t)(&Bs[buf][0]);
        v4u g0;
        g0[0] = 1u;                                   // count=1, user descriptor
        g0[1] = ldsa;                                 // lds_addr
        g0[2] = (unsigned)ga;                         // global_addr[31:0]
        g0[3] = (unsigned)(ga >> 32) | (2u << 30);    // global_addr[56:32] | type=2
        v8i g1;
        g1[0] = (2 << 16);                            // data_size = 4B
        g1[1] = (128 & 0xFFFF) << 16;                 // tensor_dim0[15:0]
        g1[2] = (32 << 16);                           // tensor_dim0[31:16]=0 | tensor_dim1[15:0]
        g1[3] = (128 << 16);                          // tensor_dim1[31:16]=0 | tile_dim0=128
        g1[4] = 32;                                   // tile_dim1=32 | tile_dim2=0
        g1[5] = 1024;                                 // tensor_dim0_stride (elements)
        g1[6] = 0;
        g1[7] = 0;
        v4i z4 = {0, 0, 0, 0};
        v8i z8 = {0, 0, 0, 0, 0, 0, 0, 0};
        __builtin_amdgcn_tensor_load_to_lds(g0, g1, z4, z4, z8, 0);
    };

    if (wv == 0) issue_tdm(0, 0);

    for (int c = 0; c < 36; ++c) {
        int r  = c >> 2;
        int ch = c & 3;
        int dy = r / 3 - 1, dx = r % 3 - 1;
        int y = py + dy, x = px + dx;
        bool inb = ((unsigned)y < 32u) && ((unsigned)x < 32u);
        const float* rowp = inb ? (fT + ((size_t)(y * 32 + x)) * CIN) : zbuf;
        const float* ap = rowp + ch * 128 + kb;

        if (wv == 0) __builtin_amdgcn_s_wait_tensorcnt(0);
        __syncthreads();
        if (wv == 0 && c + 1 < 36) issue_tdm(c + 1, (c + 1) & 1);

        const float* bsrc = &Bs[c & 1][nn * 4 + kb];
#pragma unroll 8
        for (int q = 0; q < 32; ++q) {
            v2f a  = *(const v2f*)(ap + q * 4);
            v2f bb = *(const v2f*)(bsrc + q * 128);
            acc = __builtin_amdgcn_wmma_f32_16x16x4_f32(
                false, a, false, bb, (short)0, acc, false, false);
        }
        __syncthreads();
    }

    // epilogue: bias + ReLU, store NHWC
    float bias = b1[n0 + lm];
    int rowbase = m0 + (lane >> 4) * 8;
    int col = n0 + lm;
    float* Lp = Lws + (size_t)b * NPIX * FC;
#pragma unroll
    for (int v = 0; v < 8; ++v) {
        float val = acc[v] + bias;
        val = val > 0.0f ? val : 0.0f;
        Lp[(size_t)(rowbase + v) * FC + col] = val;
    }
}

// ---------------- K3: cls/reg heads (per-pixel dot products) ----------------
__global__ void k_heads(const float* __restrict__ Lws,
                        const float* __restrict__ w_cls, const float* __restrict__ b_cls,
                        const float* __restrict__ w_reg, const float* __restrict__ b_reg,
                        float* __restrict__ clsA, float* __restrict__ regA) {
    int bid = blockIdx.x;             // b*1024 + p
    int b = bid >> 10;
    int p = bid & 1023;
    int tid = threadIdx.x;

    __shared__ float Ls[FC];
    const float* Lp = Lws + (size_t)bid * FC;
    for (int i = tid; i < FC; i += 64) Ls[i] = Lp[i];
    __syncthreads();

    if (tid < 18) {
        const float* wr = w_cls + (size_t)tid * FC;
        float acc = b_cls[tid];
#pragma unroll 8
        for (int c = 0; c < FC; ++c) acc += Ls[c] * wr[c];
        int a = tid >> 1, cc = tid & 1;
        clsA[((size_t)b * NANCH + p * ACNT + a) * 2 + cc] = acc;
    } else if (tid < 54) {
        int t2 = tid - 18;
        const float* wr = w_reg + (size_t)t2 * FC;
        float acc = b_reg[t2];
#pragma unroll 8
        for (int c = 0; c < FC; ++c) acc += Ls[c] * wr[c];
        int a = t2 >> 2, d = t2 & 3;
        regA[((size_t)b * NANCH + p * ACNT + a) * 4 + d] = acc;
    }
}

// ---------------- K4: bbox transform + softmax + sort keys ----------------
__global__ void k_boxes_keys(const float* __restrict__ clsA,
                             const float* __restrict__ regA,
                             const float* __restrict__ anchors,
                             float* __restrict__ boxA, float* __restrict__ sA,
                             int* __restrict__ lblA,
                             unsigned long long* __restrict__ keys) {
    int idx = blockIdx.x * blockDim.x + threadIdx.x;
    if (idx >= BSZ * NPAD) return;
    int b = idx >> 14;
    int n = idx & (NPAD - 1);
    unsigned long long* kk = keys + ((size_t)b << 14);
    if (n >= NANCH) { kk[n] = 0ull; return; }

    size_t gn = (size_t)b * NANCH + n;
    float c0 = clsA[gn * 2 + 0];
    float c1 = clsA[gn * 2 + 1];
    float mx = fmaxf(c0, c1);
    float e0 = expf(c0 - mx), e1 = expf(c1 - mx);
    float s = fmaxf(e0, e1) / (e0 + e1);
    int label = (c1 > c0) ? 1 : 0;

    float a0 = anchors[n * 4 + 0], a1 = anchors[n * 4 + 1];
    float a2 = anchors[n * 4 + 2], a3 = anchors[n * 4 + 3];
    float w = a2 - a0 + 1.0f, h = a3 - a1 + 1.0f;
    float cx = a0 + 0.5f * w, cy = a1 + 0.5f * h;
    float dx = regA[gn * 4 + 0], dy = regA[gn * 4 + 1];
    float dw = regA[gn * 4 + 2], dh = regA[gn * 4 + 3];
    float pcx = dx * w + cx, pcy = dy * h + cy;
    float pw = expf(dw) * w, ph = expf(dh) * h;
    boxA[gn * 4 + 0] = pcx - 0.5f * pw;
    boxA[gn * 4 + 1] = pcy - 0.5f * ph;
    boxA[gn * 4 + 2] = pcx + 0.5f * pw;
    boxA[gn * 4 + 3] = pcy + 0.5f * ph;
    sA[gn] = s;
    lblA[gn] = label;

    float sm = label ? s : -INFINITY;
    unsigned int ub = __float_as_uint(sm);
    ub = (ub & 0x80000000u) ? ~ub : (ub | 0x80000000u);
    kk[n] = ((unsigned long long)ub << 32) | (unsigned long long)(0xFFFFFFFFu - (unsigned)n);
}

// ---------------- K5: bitonic sort step (descending) ----------------
__global__ void k_bitonic_step(unsigned long long* __restrict__ keys, int j, int k) {
    int t = blockIdx.x * blockDim.x + threadIdx.x;
    if (t >= BSZ * NPAD) return;
    int b = t >> 14;
    int i = t & (NPAD - 1);
    int l = i ^ j;
    if (l > i) {
        unsigned long long* kk = keys + ((size_t)b << 14);
        unsigned long long a = kk[i], c = kk[l];
        bool up = (i & k) == 0;
        bool sw = up ? (a < c) : (a > c);
        if (sw) { kk[i] = c; kk[l] = a; }
    }
}

// ---------------- K6: gather sorted boxes/scores/labels ----------------
__global__ void k_gather_sorted(const unsigned long long* __restrict__ keys,
                                const float* __restrict__ boxA, const float* __restrict__ sA,
                                const int* __restrict__ lblA,
                                float* __restrict__ bsort, float* __restrict__ ssort,
                                int* __restrict__ lsort) {
    int t = blockIdx.x * blockDim.x + threadIdx.x;
    if (t >= BSZ * NANCH) return;
    int b = t / NANCH;
    int n = t % NANCH;
    unsigned long long key = keys[((size_t)b << 14) + n];
    unsigned idx = 0xFFFFFFFFu - (unsigned)(key & 0xFFFFFFFFull);
    size_t src = (size_t)b * NANCH + idx;
    size_t dst = (size_t)b * NANCH + n;
    bsort[dst * 4 + 0] = boxA[src * 4 + 0];
    bsort[dst * 4 + 1] = boxA[src * 4 + 1];
    bsort[dst * 4 + 2] = boxA[src * 4 + 2];
    bsort[dst * 4 + 3] = boxA[src * 4 + 3];
    ssort[dst] = sA[src];
    lsort[dst] = lblA[src];
}

// ---------------- K7: greedy NMS + output packing (one WG per image) ----------------
__global__ void k_nms_pack(const float* __restrict__ bsort, const float* __restrict__ ssort,
                           const int* __restrict__ lsort,
                           float* __restrict__ out_b, float* __restrict__ out_s,
                           int* __restrict__ out_l) {
    int b = blockIdx.x;
    int tid = threadIdx.x;
    __shared__ unsigned char keep[NANCH];

    const float* bs = bsort + (size_t)b * NANCH * 4;
    const float* ss = ssort + (size_t)b * NANCH;
    const int*   ls = lsort + (size_t)b * NANCH;

    for (int i = tid; i < NANCH; i += 256) keep[i] = (unsigned char)(ls[i] > 0);
    __syncthreads();

    int cnt = 0;
    for (int i = 0; i < NANCH && cnt < MAXOUT; ++i) {
        if (keep[i]) {
            float x1 = bs[i * 4 + 0], y1 = bs[i * 4 + 1];
            float x2 = bs[i * 4 + 2], y2 = bs[i * 4 + 3];
            float ai = (x2 - x1 + 1.0f) * (y2 - y1 + 1.0f);
            if (tid == 0) {
                size_t o = (size_t)b * MAXOUT + cnt;
                out_b[o * 4 + 0] = x1; out_b[o * 4 + 1] = y1;
                out_b[o * 4 + 2] = x2; out_b[o * 4 + 3] = y2;
                out_s[o] = ss[i];
                out_l[o] = ls[i];
            }
            for (int j = i + 1 + tid; j < NANCH; j += 256) {
                if (keep[j]) {
                    float jx1 = bs[j * 4 + 0], jy1 = bs[j * 4 + 1];
                    float jx2 = bs[j * 4 + 2], jy2 = bs[j * 4 + 3];
                    float xx1 = fmaxf(jx1, x1), yy1 = fmaxf(jy1, y1);
                    float xx2 = fminf(jx2, x2), yy2 = fminf(jy2, y2);
                    float iw = fmaxf(xx2 - xx1 + 1.0f, 0.0f);
                    float ih = fmaxf(yy2 - yy1 + 1.0f, 0.0f);
                    float inter = iw * ih;
                    float aj = (jx2 - jx1 + 1.0f) * (jy2 - jy1 + 1.0f);
                    float iou = inter / (ai + aj - inter);
                    if (iou > NMSTHR) keep[j] = 0;
                }
            }
            __syncthreads();
            cnt++;
        }
    }
    for (int slot = cnt + tid; slot < MAXOUT; slot += 256) {
        size_t o = (size_t)b * MAXOUT + slot;
        out_b[o * 4 + 0] = 0.0f; out_b[o * 4 + 1] = 0.0f;
        out_b[o * 4 + 2] = 0.0f; out_b[o * 4 + 3] = 0.0f;
        out_s[o] = -1.0f;
        out_l[o] = -1;
    }
}

extern "C" void kernel_launch(void* const* d_in, const int* in_sizes, int n_in,
                              void* d_out, int out_size, void* d_ws, size_t ws_size,
                              hipStream_t stream) {
    (void)in_sizes; (void)n_in; (void)out_size; (void)ws_size;
    const float* feature = (const float*)d_in[0];
    const float* anchors = (const float*)d_in[1];
    const float* w1      = (const float*)d_in[2];
    const float* b1      = (const float*)d_in[3];
    const float* w_cls   = (const float*)d_in[4];
    const float* b_cls   = (const float*)d_in[5];
    const float* w_reg   = (const float*)d_in[6];
    const float* b_reg   = (const float*)d_in[7];

    char* w = (char*)d_ws;
    float* featT = (float*)(w + 0);                       // 33,554,432 B
    float* zbuf  = (float*)(w + 33554432);                //      2,048 B
    float* wT2   = (float*)(w + 33556480);                //  4,718,592 B
    float* Lws   = (float*)(w + 38275072);                // 16,777,216 B
    float* clsA  = (float*)(w + 55052288);                //  1,179,648 B
    float* regA  = (float*)(w + 56231936);                //  2,359,296 B
    float* boxA  = (float*)(w + 58591232);                //  2,359,296 B
    float* sA    = (float*)(w + 60950528);                //    589,824 B
    int*   lblA  = (int*)  (w + 61540352);                //    589,824 B
    unsigned long long* keys = (unsigned long long*)(w + 62130176); // 2,097,152 B
    float* bsort = (float*)(w + 64227328);                //  2,359,296 B
    float* ssort = (float*)(w + 66586624);                //    589,824 B
    int*   lsort = (int*)  (w + 67176448);                //    589,824 B

    float* out_b = (float*)d_out;                         // 16*256*4
    float* out_s = out_b + BSZ * MAXOUT * 4;              // 16*256
    int*   out_l = (int*)(out_s + BSZ * MAXOUT);          // 16*256 (int32 bits)

    k_zero<<<1, 512, 0, stream>>>(zbuf);
    k_transpose_feat<<<(BSZ * CIN * NPIX + 255) / 256, 256, 0, stream>>>(feature, featT);
    k_repack_w<<<(9 * CIN * FC + 255) / 256, 256, 0, stream>>>(w1, wT2);
    k_conv_wmma<<<BSZ * 16 * 8, 256, 0, stream>>>(featT, zbuf, wT2, b1, Lws);
    k_heads<<<BSZ * NPIX, 64, 0, stream>>>(Lws, w_cls, b_cls, w_reg, b_reg, clsA, regA);
    k_boxes_keys<<<(BSZ * NPAD + 255) / 256, 256, 0, stream>>>(clsA, regA, anchors,
                                                              boxA, sA, lblA, keys);
    for (int k = 2; k <= NPAD; k <<= 1)
        for (int j = k >> 1; j > 0; j >>= 1)
            k_bitonic_step<<<(BSZ * NPAD + 255) / 256, 256, 0, stream>>>(keys, j, k);
    k_gather_sorted<<<(BSZ * NANCH + 255) / 256, 256, 0, stream>>>(keys, boxA, sA, lblA,
                                                                   bsort, ssort, lsort);
    k_nms_pack<<<BSZ, 256, 0, stream>>>(bsort, ssort, lsort, out_b, out_s, out_l);
}